// Net_56092272885928
// MI455X (gfx1250) — compile-verified
//
#include <hip/hip_runtime.h>
#include <hip/hip_bf16.h>
#include <cstdint>

#define BATCH 16
#define NPTS  2048
#define KNN   20

typedef float v2f __attribute__((ext_vector_type(2)));
typedef float v8f __attribute__((ext_vector_type(8)));
typedef int   v4i __attribute__((ext_vector_type(4)));

// Full-precision f32 WMMA: D(16x16) = A(16x4) * B(4x16) + C
__device__ __forceinline__ v8f wmma4(v2f a, v2f b, v8f c) {
  return __builtin_amdgcn_wmma_f32_16x16x4_f32(false, a, false, b, (short)0, c,
                                               false, false);
}

// ---------------------------------------------------------------------------
// Async global->LDS copy (gfx1250 GLOBAL_LOAD_ASYNC_TO_LDS_B128, ASYNCcnt),
// with a safe fallback if the builtin is absent on this toolchain.
// Builtin signature (from compiler diagnostic): param0 = v4i* in global AS,
// param1 = v4i* in LDS AS, then imm offset + imm cpol.
// ---------------------------------------------------------------------------
#if defined(__has_builtin)
#if __has_builtin(__builtin_amdgcn_global_load_async_to_lds_b128)
#define HAVE_ASYNC_LDS 1
#endif
#endif

#ifdef HAVE_ASYNC_LDS
__device__ __forceinline__ void async_copy16(const float* src, float* lds_dst) {
  __builtin_amdgcn_global_load_async_to_lds_b128(
      (__attribute__((address_space(1))) v4i*)(void*)src,
      (__attribute__((address_space(3))) v4i*)(void*)lds_dst, 0, 0);
}
__device__ __forceinline__ void async_wait() {
  asm volatile("s_wait_asynccnt 0x0" ::: "memory");
}
#else
__device__ __forceinline__ void async_copy16(const float* src, float* lds_dst) {
  float4 v = *(const float4*)src;
  *(float4*)lds_dst = v;
}
__device__ __forceinline__ void async_wait() {}
#endif

// ---------------------------------------------------------------------------
// kNN (k=20) over 3D positions: one wave per query point.
// Per-lane sorted top-20 (unrolled => stays in VGPRs), then 20 rounds of
// wave argmin-pop via lane shuffles.
// ---------------------------------------------------------------------------
__global__ __launch_bounds__(256) void knn3d_kernel(
    const float* __restrict__ pos, int* __restrict__ idx_out) {
  int wave = (blockIdx.x * blockDim.x + threadIdx.x) >> 5;
  int lane = threadIdx.x & 31;
  int b = wave / NPTS;
  int i = wave % NPTS;
  const float* P = pos + (size_t)b * NPTS * 3;
  float xi0 = P[i * 3 + 0], xi1 = P[i * 3 + 1], xi2 = P[i * 3 + 2];

  float bd[KNN];
  int bi[KNN];
#pragma unroll
  for (int s = 0; s < KNN; ++s) { bd[s] = 3.4e38f; bi[s] = 0; }

  for (int c = lane; c < NPTS; c += 32) {
    float d0 = P[c * 3 + 0] - xi0;
    float d1 = P[c * 3 + 1] - xi1;
    float d2 = P[c * 3 + 2] - xi2;
    float cd = d0 * d0 + d1 * d1 + d2 * d2;
    int ci = c;
#pragma unroll
    for (int s = 0; s < KNN; ++s) {
      if (cd < bd[s]) {
        float td = bd[s]; bd[s] = cd; cd = td;
        int ti = bi[s]; bi[s] = ci; ci = ti;
      }
    }
  }

  int* out = idx_out + ((size_t)b * NPTS + i) * KNN;
  for (int r = 0; r < KNN; ++r) {
    float mv = bd[0];
    int mi = bi[0];
    int ml = lane;
#pragma unroll
    for (int off = 16; off >= 1; off >>= 1) {
      float ov = __shfl_xor(mv, off, 32);
      int oi = __shfl_xor(mi, off, 32);
      int ol = __shfl_xor(ml, off, 32);
      if (ov < mv || (ov == mv && ol < ml)) { mv = ov; mi = oi; ml = ol; }
    }
    if (lane == ml) {
#pragma unroll
      for (int s = 0; s < KNN - 1; ++s) { bd[s] = bd[s + 1]; bi[s] = bi[s + 1]; }
      bd[KNN - 1] = 3.4e38f;
    }
    if (lane == 0) out[r] = mi;
  }
}

// ---------------------------------------------------------------------------
// kNN over 64-dim features using precomputed Gram matrix (one cloud).
// dist(i,c) = |f_i|^2 + |f_c|^2 - 2 <f_i, f_c>
// ---------------------------------------------------------------------------
__global__ __launch_bounds__(256) void knn_feat_kernel(
    const float* __restrict__ gram, const float* __restrict__ sq,
    int* __restrict__ idx_out, int b) {
  int wave = (blockIdx.x * blockDim.x + threadIdx.x) >> 5;  // 0..NPTS-1
  int lane = threadIdx.x & 31;
  int i = wave;
  const float* sqb = sq + (size_t)b * NPTS;
  float si = sqb[i];
  const float* grow = gram + (size_t)i * NPTS;

  float bd[KNN];
  int bi[KNN];
#pragma unroll
  for (int s = 0; s < KNN; ++s) { bd[s] = 3.4e38f; bi[s] = 0; }

  for (int c = lane; c < NPTS; c += 32) {
    float cd = si + sqb[c] - 2.0f * grow[c];
    int ci = c;
#pragma unroll
    for (int s = 0; s < KNN; ++s) {
      if (cd < bd[s]) {
        float td = bd[s]; bd[s] = cd; cd = td;
        int ti = bi[s]; bi[s] = ci; ci = ti;
      }
    }
  }

  int* out = idx_out + ((size_t)b * NPTS + i) * KNN;
  for (int r = 0; r < KNN; ++r) {
    float mv = bd[0];
    int mi = bi[0];
    int ml = lane;
#pragma unroll
    for (int off = 16; off >= 1; off >>= 1) {
      float ov = __shfl_xor(mv, off, 32);
      int oi = __shfl_xor(mi, off, 32);
      int ol = __shfl_xor(ml, off, 32);
      if (ov < mv || (ov == mv && ol < ml)) { mv = ov; mi = oi; ml = ol; }
    }
    if (lane == ml) {
#pragma unroll
      for (int s = 0; s < KNN - 1; ++s) { bd[s] = bd[s + 1]; bi[s] = bi[s + 1]; }
      bd[KNN - 1] = 3.4e38f;
    }
    if (lane == 0) out[r] = mi;
  }
}

// ---------------------------------------------------------------------------
// EdgeConv1: per query point, edge feat [xi, xj-xi] (20x6) -> 6->64 (scalar)
// -> 64->64 via WMMA -> ReLU -> max over 20 neighbors.
// One workgroup (128 thr = 4 waves) per query; wave w owns N-tile w.
// Single K-loop with two accumulators: each weight fragment loads once and
// feeds two back-to-back WMMAs.
// ---------------------------------------------------------------------------
__global__ __launch_bounds__(128) void edgeconv1_kernel(
    const float* __restrict__ pos, const int* __restrict__ idx1,
    const float* __restrict__ w1, const float* __restrict__ b1,
    const float* __restrict__ w2, const float* __restrict__ b2,
    float* __restrict__ out1) {
  int q = blockIdx.x;
  int b = q / NPTS, i = q % NPTS;
  int tid = threadIdx.x;
  __shared__ float E[KNN][6];
  __shared__ float H[32][64];
  __shared__ int nbr[KNN];
  const float* P = pos + (size_t)b * NPTS * 3;

  if (tid < KNN) nbr[tid] = idx1[((size_t)b * NPTS + i) * KNN + tid];
  __syncthreads();

  if (tid < KNN * 6) {
    int j = tid / 6, d = tid % 6;
    float v;
    if (d < 3) v = P[i * 3 + d];
    else       v = P[nbr[j] * 3 + (d - 3)] - P[i * 3 + (d - 3)];
    E[j][d] = v;
  }
  __syncthreads();

  // layer1 (K=6, scalar): H[j][c] = relu(E[j]·w1[:,c] + b1[c]); rows 20..31 = 0
  for (int t = tid; t < 32 * 64; t += 128) {
    int j = t >> 6, c = t & 63;
    float acc = 0.0f;
    if (j < KNN) {
      acc = b1[c];
#pragma unroll
      for (int d = 0; d < 6; ++d) acc += E[j][d] * w1[d * 64 + c];
      acc = fmaxf(acc, 0.0f);
    }
    H[j][c] = acc;
  }
  __syncthreads();

  // layer2 (K=64) via f32 WMMA; bias+relu; mask pad rows; max over rows
  int wv = tid >> 5, lane = tid & 31;
  int half = lane >> 4, l = lane & 15;
  int col = wv * 16 + l;  // 4 waves x 16 = 64 output channels
  v8f acc0 = {}, acc1 = {};
  for (int k = 0; k < 64; k += 4) {
    int kk = k + 2 * half;
    v2f bb; bb.x = w2[kk * 64 + col]; bb.y = w2[(kk + 1) * 64 + col];
    v2f a0; a0.x = H[l][kk];          a0.y = H[l][kk + 1];
    v2f a1; a1.x = H[16 + l][kk];     a1.y = H[16 + l][kk + 1];
    acc0 = wmma4(a0, bb, acc0);
    acc1 = wmma4(a1, bb, acc1);
  }
  float bc = b2[col];
  float cmax = -3.4e38f;
#pragma unroll
  for (int r = 0; r < 8; ++r) {
    // tile0 rows 0..15: always valid (< KNN)
    cmax = fmaxf(cmax, fmaxf(acc0[r] + bc, 0.0f));
    int row1 = 16 + r + 8 * half;
    float v1 = fmaxf(acc1[r] + bc, 0.0f);
    if (row1 >= KNN) v1 = -3.4e38f;  // mask zero-padded rows
    cmax = fmaxf(cmax, v1);
  }
  cmax = fmaxf(cmax, __shfl_xor(cmax, 16, 32));
  if (lane < 16) out1[((size_t)b * NPTS + i) * 64 + col] = cmax;
}

// ---------------------------------------------------------------------------
// |f|^2 per point (for feature-space kNN distances)
// ---------------------------------------------------------------------------
__global__ void sqnorm_kernel(const float* __restrict__ feat,
                              float* __restrict__ sq) {
  int t = blockIdx.x * blockDim.x + threadIdx.x;  // BATCH*NPTS
  const float* f = feat + (size_t)t * 64;
  float s = 0.0f;
  for (int c = 0; c < 64; ++c) { float v = f[c]; s += v * v; }
  sq[t] = s;
}

// ---------------------------------------------------------------------------
// Gram = F * F^T for one cloud, F: [2048,64] f32, via WMMA 16x16x4.
// One wave per 16x16 output tile. All 8 waves of a block share the same
// 16-row A tile (contiguous 4KB) -> staged once via async global->LDS copy.
// ---------------------------------------------------------------------------
__global__ __launch_bounds__(256) void gram_kernel(
    const float* __restrict__ feat, float* __restrict__ gram, int b) {
  __shared__ float As[16 * 64];  // 4 KB staged A tile
  int tid = threadIdx.x;
  int wv = tid >> 5, lane = tid & 31, half = lane >> 4, l = lane & 15;
  int t0 = blockIdx.x * 8;       // first tile id of this block
  int mt = t0 >> 7;              // same mt for all 8 waves (128 % 8 == 0)
  int nt = (t0 & 127) + wv;
  const float* F = feat + (size_t)b * NPTS * 64;

  // stage A tile: 1024 floats, 16B per thread, async (ASYNCcnt) if available
  async_copy16(F + (size_t)mt * 16 * 64 + tid * 4, &As[tid * 4]);
  async_wait();
  __syncthreads();

  const float* Brow = F + (size_t)(nt * 16 + l) * 64;
  v8f acc = {};
  for (int k = 0; k < 64; k += 4) {
    int kk = k + 2 * half;
    v2f a; a.x = As[l * 64 + kk]; a.y = As[l * 64 + kk + 1];
    v2f bb; bb.x = Brow[kk];      bb.y = Brow[kk + 1];
    acc = wmma4(a, bb, acc);
  }
#pragma unroll
  for (int r = 0; r < 8; ++r) {
    gram[(size_t)(mt * 16 + r + 8 * half) * NPTS + nt * 16 + l] = acc[r];
  }
}

// ---------------------------------------------------------------------------
// EdgeConv2: per query, edge feat [fi, fj-fi] (20x128) @ W (128x128) + b,
// ReLU, max over 20. One workgroup (256 thr = 8 waves) per query.
// Neighbor rows are gathered into LDS via async global->LDS B128 copies.
// ---------------------------------------------------------------------------
__global__ __launch_bounds__(256) void edgeconv2_kernel(
    const float* __restrict__ feat, const int* __restrict__ idx2,
    const float* __restrict__ w, const float* __restrict__ bias,
    float* __restrict__ out2) {
  int q = blockIdx.x;
  int b = q / NPTS, i = q % NPTS;
  int tid = threadIdx.x;
  __shared__ float Raw[(KNN + 1) * 64];  // row 0 = f_i, rows 1..20 = neighbors
  __shared__ float E[32][128];           // 16 KB edge-feature matrix
  __shared__ int nbr[KNN];
  const float* F = feat + (size_t)b * NPTS * 64;

  if (tid < KNN) nbr[tid] = idx2[((size_t)b * NPTS + i) * KNN + tid];
  __syncthreads();

  // async gather: 21 rows x 16 chunks of 16B
  for (int c = tid; c < (KNN + 1) * 16; c += 256) {
    int row = c >> 4, ch = (c & 15) * 4;
    int srow = (row == 0) ? i : nbr[row - 1];
    async_copy16(F + (size_t)srow * 64 + ch, &Raw[row * 64 + ch]);
  }
  async_wait();
  __syncthreads();

  // build E[j][0..63] = f_i ; E[j][64..127] = f_j - f_i ; pad rows = 0
  for (int t = tid; t < 32 * 128; t += 256) {
    int j = t >> 7, c = t & 127;
    float v = 0.0f;
    if (j < KNN) {
      float fi = Raw[c & 63];
      v = (c < 64) ? fi : (Raw[(j + 1) * 64 + (c - 64)] - fi);
    }
    E[j][c] = v;
  }
  __syncthreads();

  int wv = tid >> 5, lane = tid & 31, half = lane >> 4, l = lane & 15;
  int col = wv * 16 + l;  // 8 waves x 16 = 128 output channels
  v8f acc0 = {}, acc1 = {};
  for (int k = 0; k < 128; k += 4) {
    int kk = k + 2 * half;
    v2f bb; bb.x = w[kk * 128 + col]; bb.y = w[(kk + 1) * 128 + col];
    v2f a0; a0.x = E[l][kk];          a0.y = E[l][kk + 1];
    v2f a1; a1.x = E[16 + l][kk];     a1.y = E[16 + l][kk + 1];
    acc0 = wmma4(a0, bb, acc0);
    acc1 = wmma4(a1, bb, acc1);
  }
  float bc = bias[col];
  float cmax = -3.4e38f;
#pragma unroll
  for (int r = 0; r < 8; ++r) {
    cmax = fmaxf(cmax, fmaxf(acc0[r] + bc, 0.0f));  // rows 0..15 valid
    int row1 = 16 + r + 8 * half;
    float v1 = fmaxf(acc1[r] + bc, 0.0f);
    if (row1 >= KNN) v1 = -3.4e38f;
    cmax = fmaxf(cmax, v1);
  }
  cmax = fmaxf(cmax, __shfl_xor(cmax, 16, 32));
  if (lane < 16) out2[((size_t)b * NPTS + i) * 128 + col] = cmax;
}

// ---------------------------------------------------------------------------
// m1: relu([out1|out2] @ W(192x512) + b), summed over 16-row tiles into
// per-tile partial sums (deterministic mean pool, no atomics).
// One wave per (b, col-tile, row-tile).
// ---------------------------------------------------------------------------
__global__ __launch_bounds__(256) void m1_kernel(
    const float* __restrict__ out1, const float* __restrict__ out2,
    const float* __restrict__ w, const float* __restrict__ bias,
    float* __restrict__ psum) {
  int gw = (blockIdx.x * blockDim.x + threadIdx.x) >> 5;  // b*4096+ct*128+rt
  int lane = threadIdx.x & 31, half = lane >> 4, l = lane & 15;
  int rt = gw & 127;
  int ct = (gw >> 7) & 31;
  int b = gw >> 12;
  int col = ct * 16 + l;
  const float* F1 = out1 + (size_t)b * NPTS * 64;
  const float* F2 = out2 + (size_t)b * NPTS * 128;
  int row = rt * 16 + l;

  v8f acc = {};
  for (int k = 0; k < 192; k += 4) {
    int kk = k + 2 * half;
    v2f a;
    if (kk < 64) { a.x = F1[(size_t)row * 64 + kk];         a.y = F1[(size_t)row * 64 + kk + 1]; }
    else         { a.x = F2[(size_t)row * 128 + (kk - 64)]; a.y = F2[(size_t)row * 128 + (kk - 63)]; }
    v2f bb; bb.x = w[kk * 512 + col]; bb.y = w[(kk + 1) * 512 + col];
    acc = wmma4(a, bb, acc);
  }
  float bc = bias[col];
  float s = 0.0f;
#pragma unroll
  for (int r = 0; r < 8; ++r) s += fmaxf(acc[r] + bc, 0.0f);
  s += __shfl_xor(s, 16, 32);  // rows 0-7 + rows 8-15 of the tile
  if (lane < 16) psum[((size_t)b * 512 + col) * 128 + rt] = s;
}

__global__ void mean_kernel(const float* __restrict__ psum,
                            float* __restrict__ g) {
  int t = blockIdx.x * blockDim.x + threadIdx.x;  // BATCH*512
  const float* p = psum + (size_t)t * 128;
  float s = 0.0f;
  for (int r = 0; r < 128; ++r) s += p[r];
  g[t] = s * (1.0f / NPTS);
}

// ---------------------------------------------------------------------------
// Tiny tail FC layers (deterministic scalar dot products)
// ---------------------------------------------------------------------------
__global__ void fc_kernel(const float* __restrict__ in,
                          const float* __restrict__ w,
                          const float* __restrict__ bias,
                          float* __restrict__ out, int IN, int OUT, int relu) {
  int t = blockIdx.x * blockDim.x + threadIdx.x;
  if (t >= BATCH * OUT) return;
  int b = t / OUT, o = t % OUT;
  const float* x = in + (size_t)b * IN;
  float s = bias[o];
  for (int k = 0; k < IN; ++k) s += x[k] * w[k * OUT + o];
  out[t] = relu ? fmaxf(s, 0.0f) : s;
}

// ---------------------------------------------------------------------------

extern "C" void kernel_launch(void* const* d_in, const int* in_sizes, int n_in,
                              void* d_out, int out_size, void* d_ws,
                              size_t ws_size, hipStream_t stream) {
  const float* pos   = (const float*)d_in[0];
  const float* c1_w1 = (const float*)d_in[1];
  const float* c1_b1 = (const float*)d_in[2];
  const float* c1_w2 = (const float*)d_in[3];
  const float* c1_b2 = (const float*)d_in[4];
  const float* c2_w1 = (const float*)d_in[5];
  const float* c2_b1 = (const float*)d_in[6];
  const float* m1_w1 = (const float*)d_in[7];
  const float* m1_b1 = (const float*)d_in[8];
  const float* m2_w1 = (const float*)d_in[9];
  const float* m2_b1 = (const float*)d_in[10];
  const float* m2_w2 = (const float*)d_in[11];
  const float* m2_b2 = (const float*)d_in[12];
  const float* m2_w3 = (const float*)d_in[13];
  const float* m2_b3 = (const float*)d_in[14];
  float* out = (float*)d_out;

  // workspace carve-up (aligned to 256B)
  char* p = (char*)d_ws;
  auto carve = [&](size_t bytes) {
    char* r = p;
    p += (bytes + 255) & ~(size_t)255;
    return (void*)r;
  };
  int*   idx1 = (int*)carve((size_t)BATCH * NPTS * KNN * 4);
  int*   idx2 = (int*)carve((size_t)BATCH * NPTS * KNN * 4);
  float* out1 = (float*)carve((size_t)BATCH * NPTS * 64 * 4);
  float* out2 = (float*)carve((size_t)BATCH * NPTS * 128 * 4);
  float* sq2  = (float*)carve((size_t)BATCH * NPTS * 4);
  float* gram = (float*)carve((size_t)NPTS * NPTS * 4);  // reused per cloud
  float* psum = (float*)carve((size_t)BATCH * 512 * 128 * 4);
  float* g    = (float*)carve((size_t)BATCH * 512 * 4);
  float* g2   = (float*)carve((size_t)BATCH * 512 * 4);
  float* g3   = (float*)carve((size_t)BATCH * 256 * 4);

  // 1) kNN in 3D (all clouds): 16*2048 waves, 8 waves/block
  knn3d_kernel<<<(BATCH * NPTS) / 8, 256, 0, stream>>>(pos, idx1);

  // 2) EdgeConv1: one workgroup per query
  edgeconv1_kernel<<<BATCH * NPTS, 128, 0, stream>>>(pos, idx1, c1_w1, c1_b1,
                                                     c1_w2, c1_b2, out1);

  // 3) squared norms of 64-dim features
  sqnorm_kernel<<<(BATCH * NPTS) / 256, 256, 0, stream>>>(out1, sq2);

  // 4) per-cloud Gram (WMMA) + feature-space kNN (stream-ordered reuse)
  for (int b = 0; b < BATCH; ++b) {
    gram_kernel<<<(128 * 128) / 8, 256, 0, stream>>>(out1, gram, b);
    knn_feat_kernel<<<NPTS / 8, 256, 0, stream>>>(gram, sq2, idx2, b);
  }

  // 5) EdgeConv2: one workgroup per query
  edgeconv2_kernel<<<BATCH * NPTS, 256, 0, stream>>>(out1, idx2, c2_w1, c2_b1,
                                                     out2);

  // 6) m1 GEMM + partial row sums, then deterministic mean
  m1_kernel<<<(BATCH * 32 * 128) / 8, 256, 0, stream>>>(out1, out2, m1_w1,
                                                        m1_b1, psum);
  mean_kernel<<<(BATCH * 512) / 256, 256, 0, stream>>>(psum, g);

  // 7) tail MLP
  fc_kernel<<<(BATCH * 512 + 255) / 256, 256, 0, stream>>>(g, m2_w1, m2_b1, g2,
                                                           512, 512, 1);
  fc_kernel<<<(BATCH * 256 + 255) / 256, 256, 0, stream>>>(g2, m2_w2, m2_b2,
                                                           g3, 512, 256, 1);
  fc_kernel<<<1, 160, 0, stream>>>(g3, m2_w3, m2_b3, out, 256, 10, 0);
}